// HighPassEncoder_46377056862934
// MI455X (gfx1250) — compile-verified
//
#include <hip/hip_runtime.h>
#include <hip/hip_bf16.h>

// HighPassEncoder: out = relu((x - deg_inv[col]*scatter_sum(x[row] -> col)) @ W^T + b)
// N=50000, E=800000, IN=OUT=128, fp32 end-to-end (matches reference precision).

typedef float v2f __attribute__((ext_vector_type(2)));
typedef float v8f __attribute__((ext_vector_type(8)));

#define DIM 128
#define HSTRIDE 132  // LDS row stride (pad to avoid 16-way bank conflicts)

// ---------------- 1) zero agg (d_out) + deg (ws) ----------------
__global__ void hp_zero(float* __restrict__ agg, float* __restrict__ deg,
                        int total_out, int n) {
    int i = blockIdx.x * blockDim.x + threadIdx.x;
    int stride = gridDim.x * blockDim.x;
    for (int k = i; k < total_out; k += stride) agg[k] = 0.0f;
    for (int k = i; k < n; k += stride) deg[k] = 0.0f;
}

// ---------------- 2) in-degree of destination nodes ----------------
__global__ void hp_degree(const int* __restrict__ ei, float* __restrict__ deg, int E) {
    int e = blockIdx.x * blockDim.x + threadIdx.x;
    if (e < E) {
        int col = ei[E + e];  // edge_index[1][e]
        atomicAdd(&deg[col], 1.0f);
    }
}

// ---------------- 3) scatter: agg[col] += x[row]  (one wave per edge) ----------------
__global__ void hp_scatter(const float* __restrict__ x, const int* __restrict__ ei,
                           float* __restrict__ agg, int E) {
    int lane  = threadIdx.x & 31;
    int wave  = (blockIdx.x * blockDim.x + threadIdx.x) >> 5;
    int nwave = (gridDim.x * blockDim.x) >> 5;
    for (int e = wave; e < E; e += nwave) {
        int row = ei[e];
        int col = ei[E + e];
        const float4 v = *(const float4*)(x + (size_t)row * DIM + lane * 4);
        float* dst = agg + (size_t)col * DIM + lane * 4;
        atomicAdd(dst + 0, v.x);
        atomicAdd(dst + 1, v.y);
        atomicAdd(dst + 2, v.z);
        atomicAdd(dst + 3, v.w);
    }
}

// ---------------- 4) h = x - deg_inv*agg ; out = relu(h @ W^T + b)  ----------------
// One block per 16-node tile; 8 waves, wave w owns output columns [16w, 16w+16).
// Uses V_WMMA_F32_16X16X4_F32, K-loop of 32 steps (K=128), fp32 accumulate.
__global__ __launch_bounds__(256) void hp_wmma(const float* __restrict__ x,
                                               const float* __restrict__ W,
                                               const float* __restrict__ bias,
                                               const float* __restrict__ deg,
                                               float* __restrict__ out /* in: agg, out: result */,
                                               int n) {
    __shared__ float hTile[16 * HSTRIDE];  // 8.25 KB

    const int row0 = blockIdx.x * 16;  // n % 16 == 0 for this problem (50000/16=3125)
    const int tid  = threadIdx.x;

    // Stage h tile: 16 rows x 128 cols, 2048 elements / 256 threads = 8 each.
    for (int idx = tid; idx < 16 * DIM; idx += 256) {
        int r = idx >> 7;       // 0..15
        int k = idx & (DIM - 1);
        int node = row0 + r;
        float d = deg[node];
        float dinv = (d > 0.0f) ? (1.0f / d) : 0.0f;
        size_t off = (size_t)node * DIM + k;
        hTile[r * HSTRIDE + k] = x[off] - dinv * out[off];
    }
    __syncthreads();

    const int wave = tid >> 5;          // 0..7 -> output column tile
    const int lane = tid & 31;
    const int half = lane >> 4;         // 0: K pair {0,1}; 1: K pair {2,3}
    const int l16  = lane & 15;

    const int jcol = wave * 16 + l16;               // output column (and W row)
    const float* wrow = W + (size_t)jcol * DIM;     // W[j][*] row-major

    v8f c = {};
    #pragma unroll
    for (int kk = 0; kk < DIM; kk += 4) {
        int ka = kk + half * 2;
        // A fragment: lane (l16, half) holds h[M=l16][ka], h[M=l16][ka+1]
        v2f a = *(const v2f*)&hTile[l16 * HSTRIDE + ka];
        // B fragment: lane (l16, half) holds W^T[K=ka][N=l16], W^T[K=ka+1][N=l16]
        //           = W[jcol][ka], W[jcol][ka+1] (contiguous in W row)
        v2f b = *(const v2f*)&wrow[ka];
        c = __builtin_amdgcn_wmma_f32_16x16x4_f32(
                /*neg_a=*/false, a, /*neg_b=*/false, b,
                /*c_mod=*/(short)0, c, /*reuse_a=*/false, /*reuse_b=*/false);
    }

    const float bj = bias[jcol];
    // D layout: VGPR r, lanes 0-15 -> M=r, lanes 16-31 -> M=r+8; N = l16.
    #pragma unroll
    for (int r = 0; r < 8; ++r) {
        float v = c[r] + bj;
        v = v > 0.0f ? v : 0.0f;
        out[(size_t)(row0 + r + half * 8) * DIM + jcol] = v;
    }
}

extern "C" void kernel_launch(void* const* d_in, const int* in_sizes, int n_in,
                              void* d_out, int out_size, void* d_ws, size_t ws_size,
                              hipStream_t stream) {
    const float* x    = (const float*)d_in[0];
    const int*   ei   = (const int*)d_in[1];
    const float* W    = (const float*)d_in[2];
    const float* bias = (const float*)d_in[3];
    float* out = (float*)d_out;

    const int n = in_sizes[0] / DIM;   // 50000
    const int E = in_sizes[1] / 2;     // 800000

    float* deg = (float*)d_ws;         // N floats (200 KB) of scratch

    // 1) zero aggregation buffer (d_out) + degree buffer
    hp_zero<<<1024, 256, 0, stream>>>(out, deg, n * DIM, n);

    // 2) in-degrees
    hp_degree<<<(E + 255) / 256, 256, 0, stream>>>(ei, deg, E);

    // 3) scatter-sum of neighbor features (one wave per edge, float4 per lane)
    hp_scatter<<<2048, 256, 0, stream>>>(x, ei, out, E);

    // 4) high-pass + linear + relu via fp32 WMMA (16x16x4)
    hp_wmma<<<n / 16, 256, 0, stream>>>(x, W, bias, deg, out, n);
}